// PrototypeLoss_57853209477371
// MI455X (gfx1250) — compile-verified
//
#include <hip/hip_runtime.h>

typedef __attribute__((ext_vector_type(2))) float v2f;
typedef __attribute__((ext_vector_type(4))) float v4f;
typedef __attribute__((ext_vector_type(8))) float v8f;

#define B_DIM 2048
#define C_DIM 80
#define D_DIM 1024
#define NROWS (B_DIM * C_DIM)   // 163840
#define GRID  4096              // persistent blocks; 40 rows each
#define NACC  64                // spread accumulator cells

__global__ void init_acc_kernel(double* acc) {
    const int i = threadIdx.x;
    if (i < NACC) acc[i] = 0.0;
}

// Persistent blocks: each of the GRID blocks walks NROWS/GRID rows with a
// grid stride. Masked-out rows (label != 1) are skipped uniformly (block-wide
// branch keeps EXEC all-1s around the WMMAs and saves their 4KB of HBM
// traffic). Squared diffs accumulate directly in the WMMA C accumulator
// across all rows; the LDS/wave reduction and the f64 atomic happen once per
// block, not once per row.
__global__ __launch_bounds__(256) void row_sqdist_kernel(
    const float* __restrict__ proto,    // [C, D]
    const float* __restrict__ feat,     // [B, C, D]
    const int*   __restrict__ labels,   // [B, C]
    double*      __restrict__ acc)      // [NACC]
{
    const int tid = threadIdx.x;

    v2f ones; ones[0] = 1.0f; ones[1] = 1.0f;
    v8f cacc = {};

    for (int bc = blockIdx.x; bc < NROWS; bc += GRID) {
        // Uniform per block -> scalar load + uniform branch (EXEC stays all-1s).
        if (labels[bc] == 1) {
            const int c = bc % C_DIM;
            const v4f* frow = (const v4f*)(feat  + (size_t)bc * D_DIM);
            const v4f* prow = (const v4f*)(proto + (size_t)c  * D_DIM);

            // Feature row is read-once: stream nontemporally. Prototype rows
            // (327 KB total) stay resident in the 192 MB L2.
            v4f f = __builtin_nontemporal_load(&frow[tid]);
            v4f p = prow[tid];

            float dx = f.x - p.x;
            float dy = f.y - p.y;
            float dz = f.z - p.z;
            float dw = f.w - p.w;

            v2f a0; a0[0] = dx * dx; a0[1] = dy * dy;
            v2f a1; a1[0] = dz * dz; a1[1] = dw * dw;

            // D = A x ones + C : accumulates rowsum(A) replicated over 16
            // columns; total over all C entries = 16 * sum(A).
            cacc = __builtin_amdgcn_wmma_f32_16x16x4_f32(
                false, a0, false, ones, (short)0, cacc, false, false);
            cacc = __builtin_amdgcn_wmma_f32_16x16x4_f32(
                false, a1, false, ones, (short)0, cacc, false, false);
        }
    }

    // Per-lane total of the 8 accumulator rows.
    float s = cacc[0] + cacc[1] + cacc[2] + cacc[3]
            + cacc[4] + cacc[5] + cacc[6] + cacc[7];

    // wave32 butterfly reduction (lane-permute hw, no LDS, no barriers).
    #pragma unroll
    for (int m = 16; m >= 1; m >>= 1) {
        s += __shfl_xor(s, m, 32);
    }

    __shared__ float wsum[8];
    if ((tid & 31) == 0) wsum[tid >> 5] = s;
    __syncthreads();

    if (tid == 0) {
        float b = wsum[0] + wsum[1] + wsum[2] + wsum[3]
                + wsum[4] + wsum[5] + wsum[6] + wsum[7];
        // 1/16 undoes the 16 replicated WMMA columns (exact, power of two).
        atomicAdd(&acc[blockIdx.x & (NACC - 1)], (double)(b * 0.0625f));
    }
}

__global__ __launch_bounds__(256) void finalize_kernel(
    const int*    __restrict__ labels,
    const double* __restrict__ acc,
    float*        __restrict__ out)
{
    __shared__ int red[256];
    const int tid = threadIdx.x;
    int cnt = 0;
    for (int i = tid; i < NROWS; i += 256) {
        cnt += (labels[i] == 1) ? 1 : 0;
    }
    red[tid] = cnt;
    __syncthreads();
    #pragma unroll
    for (int off = 128; off > 0; off >>= 1) {
        if (tid < off) red[tid] += red[tid + off];
        __syncthreads();
    }
    if (tid == 0) {
        double total = 0.0;
        for (int i = 0; i < NACC; ++i) total += acc[i];
        const int count = red[0];
        out[0] = (count > 0) ? (float)(total / (double)count) : 0.0f;
    }
}

extern "C" void kernel_launch(void* const* d_in, const int* in_sizes, int n_in,
                              void* d_out, int out_size, void* d_ws, size_t ws_size,
                              hipStream_t stream) {
    const float* proto  = (const float*)d_in[0];  // class_prototype [C, D]
    const float* feat   = (const float*)d_in[1];  // feature_proj   [B, C, D]
    const int*   labels = (const int*)d_in[2];    // labels         [B, C]
    float*       out    = (float*)d_out;
    double*      acc    = (double*)d_ws;          // NACC doubles of scratch

    init_acc_kernel<<<1, NACC, 0, stream>>>(acc);
    row_sqdist_kernel<<<GRID, 256, 0, stream>>>(proto, feat, labels, acc);
    finalize_kernel<<<1, 256, 0, stream>>>(labels, acc, out);
}